// Qwen3NextGatedDeltaNetDecode_48009144435151
// MI455X (gfx1250) — compile-verified
//
#include <hip/hip_runtime.h>
#include <cstddef>

typedef __attribute__((ext_vector_type(2))) float v2f;
typedef __attribute__((ext_vector_type(8))) float v8f;

#define B_    256
#define HK_   16
#define HV_   32
#define DK_   128
#define DV_   128
#define QKV_  8192
#define EPS_  1e-6f
#define INV_SQRT_DK_ 0.08838834764831845f  // 1/sqrt(128)

// One workgroup (8 wave32) per (b, h). Streams the 128x128 fp32 state once
// (non-temporal), folding both GEMV reductions (q^T S and k^T S) into
// V_WMMA_F32_16X16X4_F32 accumulations (C rows 0 and 1).
//
//   o[v] = eg*(q^T S)[v] + (q.k)*beta*( v[v] - eg*(k^T S)[v] )
__global__ __launch_bounds__(256) void gdn_decode_kernel(
    const float* __restrict__ mixed_qkv,   // (B, QKV)
    const float* __restrict__ a_in,        // (B, HV)
    const float* __restrict__ b_in,        // (B, HV)
    const float* __restrict__ conv_state,  // (B, QKV, 3)
    const float* __restrict__ conv_w,      // (QKV, 4)
    const float* __restrict__ dt_bias,     // (HV)
    const float* __restrict__ alog,        // (HV)
    const float* __restrict__ ssm,         // (B, HV, DK, DV)
    float* __restrict__ out)               // (B, HV, DV)
{
  // afrag[0] = q_hat, afrag[1] = k_hat, afrag[2] = zeros (A-matrix rows 2..15)
  __shared__ float afrag[3][DK_];
  __shared__ float lv[DV_];
  __shared__ float red[3][DK_];
  __shared__ float part[3][32];
  __shared__ float scal[5];   // 0:rq 1:rk 2:qkdot 3:eg 4:beta

  const int blk  = blockIdx.x;
  const int b    = blk >> 5;        // / HV_
  const int h    = blk & (HV_ - 1);
  const int tid  = threadIdx.x;
  const int lane = tid & 31;
  const int w    = tid >> 5;        // wave id 0..7

  // ---------------- prologue: conv + SiLU into LDS ----------------
  if (tid < DK_) {
    const int d  = tid;
    const int qh = h >> 1;                       // GQA: 2 v-heads per k-head
    const int cq = qh * DK_ + d;                 // q channels [0, 2048)
    const int ck = HK_ * DK_ + qh * DK_ + d;     // k channels [2048, 4096)
    const int cv = 2 * HK_ * DK_ + h * DK_ + d;  // v channels [4096, 8192)

    auto conv1 = [&](int c) -> float {
      const float* cs = conv_state + ((size_t)b * QKV_ + c) * 3;
      const float* wp = conv_w + (size_t)c * 4;
      float s = cs[0] * wp[0] + cs[1] * wp[1] + cs[2] * wp[2]
              + mixed_qkv[(size_t)b * QKV_ + c] * wp[3];
      return s / (1.0f + expf(-s));              // SiLU
    };

    const float qv = conv1(cq);
    const float kv = conv1(ck);
    const float vv = conv1(cv);
    afrag[0][d] = qv;
    afrag[1][d] = kv;
    afrag[2][d] = 0.f;                           // zero A rows 2..15
    lv[d] = vv;
    red[0][d] = qv * qv;
    red[1][d] = kv * kv;
    red[2][d] = qv * kv;
  }
  __syncthreads();

  if (tid < 32) {
    #pragma unroll
    for (int j = 0; j < 3; ++j)
      part[j][tid] = red[j][tid] + red[j][tid + 32] + red[j][tid + 64] + red[j][tid + 96];
  }
  __syncthreads();

  if (tid == 0) {
    float sq = 0.f, sk = 0.f, sqk = 0.f;
    #pragma unroll
    for (int i = 0; i < 32; ++i) { sq += part[0][i]; sk += part[1][i]; sqk += part[2][i]; }
    const float rq = rsqrtf(sq + EPS_) * INV_SQRT_DK_;   // norm + DK^-0.5
    const float rk = rsqrtf(sk + EPS_);
    scal[0] = rq;
    scal[1] = rk;
    scal[2] = sqk * rq * rk;                             // q_hat . k_hat
    const float av = a_in[b * HV_ + h] + dt_bias[h];
    const float sp = (av > 20.f) ? av : log1pf(expf(av));  // softplus
    scal[3] = expf(-expf(alog[h]) * sp);                 // eg = exp(g)
    scal[4] = 1.f / (1.f + expf(-b_in[b * HV_ + h]));    // beta = sigmoid(b)
  }
  __syncthreads();

  if (tid < DK_) {               // normalize q,k in place
    afrag[0][tid] *= scal[0];
    afrag[1][tid] *= scal[1];
  }
  __syncthreads();

  const float eg   = scal[3];
  const float beta = scal[4];
  const float qk   = scal[2];

  // ---------------- main loop: stream S, WMMA-accumulate ----------------
  // Wave w owns columns [16w, 16w+16). K in tiles of 4 (32 iterations).
  // A (16x4 f32): lanes0-15 = rows M with K0/K1 in VGPR0/1; lanes16-31 = K2/K3.
  //               Row0 = q_hat, Row1 = k_hat, rows 2..15 = 0 (zero LDS row).
  // B (4x16 f32): VGPR0 = row K0 (lanes0-15) / K2 (lanes16-31); VGPR1 = K1/K3.
  // C (16x16 f32): VGPR j = row j on lanes0-15 -> row0 = q^T S, row1 = k^T S.
  const size_t sbase = ((size_t)b * HV_ + h) * (size_t)(DK_ * DV_);
  const int col  = (w << 4) + (lane & 15);
  const int rsel = (lane >> 4) << 1;          // 0 (lanes 0-15) or 2 (lanes 16-31)
  const int sel  = lane & 15;
  const int asel = (sel < 2) ? sel : 2;       // 0->q, 1->k, else zero row

  // Branch-free per-lane bases: in-loop addressing is pure immediate offsets.
  const float* __restrict__ abase = &afrag[asel][rsel];       // +4*t floats/iter
  const float* __restrict__ gbase = ssm + sbase + (size_t)rsel * DV_ + col;

  v8f acc = {0.f, 0.f, 0.f, 0.f, 0.f, 0.f, 0.f, 0.f};

  #pragma unroll 8
  for (int t = 0; t < 32; ++t) {
    const float* p = gbase + (size_t)t * (4 * DV_);
    const float b0 = __builtin_nontemporal_load(p);        // S[4t+rsel  ][col]
    const float b1 = __builtin_nontemporal_load(p + DV_);  // S[4t+rsel+1][col]
    const v2f A = *(const v2f*)(abase + 4 * t);            // ds_load_b64 imm ofs
    v2f Bf; Bf.x = b0; Bf.y = b1;
    acc = __builtin_amdgcn_wmma_f32_16x16x4_f32(
        /*neg_a=*/false, A, /*neg_b=*/false, Bf,
        /*c_mod=*/(short)0, acc, /*reuse_a=*/false, /*reuse_b=*/false);
  }

  // ---------------- epilogue ----------------
  if (lane < 16) {
    const int v    = (w << 4) + lane;
    const float qS = acc[0];           // C row 0
    const float kS = acc[1];           // C row 1
    const float o  = eg * qS + qk * beta * (lv[v] - eg * kS);
    out[((size_t)b * HV_ + h) * DV_ + v] = o;
  }
}

extern "C" void kernel_launch(void* const* d_in, const int* in_sizes, int n_in,
                              void* d_out, int out_size, void* d_ws, size_t ws_size,
                              hipStream_t stream) {
  (void)in_sizes; (void)n_in; (void)out_size; (void)d_ws; (void)ws_size;
  const float* mixed_qkv  = (const float*)d_in[0];
  const float* a_in       = (const float*)d_in[1];
  const float* b_in       = (const float*)d_in[2];
  const float* conv_state = (const float*)d_in[3];
  const float* conv_w     = (const float*)d_in[4];
  const float* dt_bias    = (const float*)d_in[5];
  const float* alog       = (const float*)d_in[6];
  const float* ssm        = (const float*)d_in[7];
  float* out = (float*)d_out;

  gdn_decode_kernel<<<dim3(B_ * HV_), dim3(256), 0, stream>>>(
      mixed_qkv, a_in, b_in, conv_state, conv_w, dt_bias, alog, ssm, out);
}